// GATNet_67405216744282
// MI455X (gfx1250) — compile-verified
//
#include <hip/hip_runtime.h>
#include <hip/hip_bf16.h>

typedef __attribute__((ext_vector_type(2))) float v2f;
typedef __attribute__((ext_vector_type(8))) float v8f;

#define N_NODES 20000
#define N_EDGES 320000
#define T_DIM   8
#define C_DIM   32
#define M_ROWS  (N_NODES * T_DIM)   /* 160000 flattened (node,t) rows */
#define NTC     (M_ROWS * C_DIM)    /* 5,120,000 elements per feature map */

// ---------------------------------------------------------------------------
// GEMM: out[M,32] = A[M,32] @ W[32,32], exact fp32 via V_WMMA_F32_16X16X4_F32.
// One wave handles a 16-row tile; K=32 -> 8 WMMA steps x 2 column halves.
// A (16x4 fp32) layout: lane L -> row = L&15, VGPR v -> K = v + 2*(L>>4).
// B (4x16  fp32) layout: lane L -> col = L&15, VGPR v -> K-row = v + 2*(L>>4).
// D (16x16 fp32) layout: VGPR j, lane L -> row = j + 8*(L>>4), col = L&15.
// ---------------------------------------------------------------------------
__global__ __launch_bounds__(128) void gat_gemm32(const float* __restrict__ A,
                                                  const float* __restrict__ W,
                                                  float* __restrict__ out, int M) {
    const int wave = threadIdx.x >> 5;
    const int lane = threadIdx.x & 31;
    const int tile = blockIdx.x * 4 + wave;
    const int m0   = tile * 16;
    if (m0 >= M) return;

    const int half = lane >> 4;   // which 16-lane half
    const int l15  = lane & 15;
    const float* __restrict__ arow = A + (size_t)(m0 + l15) * C_DIM;

    v8f acc0 = {};
    v8f acc1 = {};
#pragma unroll
    for (int k = 0; k < 8; ++k) {
        const int kk = 4 * k + 2 * half;        // even -> 8B aligned pair
        v2f a;
        a.x = arow[kk + 0];
        a.y = arow[kk + 1];
        v2f b0, b1;
        b0.x = W[(kk + 0) * C_DIM + l15];
        b0.y = W[(kk + 1) * C_DIM + l15];
        b1.x = W[(kk + 0) * C_DIM + 16 + l15];
        b1.y = W[(kk + 1) * C_DIM + 16 + l15];
        acc0 = __builtin_amdgcn_wmma_f32_16x16x4_f32(false, a, false, b0,
                                                     (short)0, acc0, false, false);
        acc1 = __builtin_amdgcn_wmma_f32_16x16x4_f32(false, a, false, b1,
                                                     (short)0, acc1, false, false);
    }
#pragma unroll
    for (int j = 0; j < 8; ++j) {
        const size_t row = (size_t)(m0 + j + 8 * half);
        out[row * C_DIM + l15]      = acc0[j];
        out[row * C_DIM + 16 + l15] = acc1[j];
    }
}

// ---------------------------------------------------------------------------
// Monotonic float -> uint encoding so GLOBAL_ATOMIC_MAX_U32 implements
// float max (order-independent => deterministic).
//   f >= 0 : u = bits | 0x80000000
//   f <  0 : u = ~bits
// Initial value 0 decodes to NaN -> squashed by the finite-check in combine,
// matching the reference's "empty segment / non-finite -> 0" semantics.
// ---------------------------------------------------------------------------
__device__ __forceinline__ unsigned enc_f32(float f) {
    unsigned u = __float_as_uint(f);
    return (u & 0x80000000u) ? ~u : (u | 0x80000000u);
}

__global__ void gat_init_agg(unsigned* __restrict__ agg, int n) {
    int i = blockIdx.x * blockDim.x + threadIdx.x;
    if (i < n) agg[i] = 0u;
}

// One thread per (edge, t): gather hu[src] & hu[tgt] rows (128B each),
// dot over 32 channels, sigmoid gate, 32 encoded atomic-max into agg[tgt].
__global__ __launch_bounds__(256) void gat_edge(const float* __restrict__ hu,
                                                const int* __restrict__ src,
                                                const int* __restrict__ tgt,
                                                unsigned* __restrict__ agg) {
    const int idx = blockIdx.x * blockDim.x + threadIdx.x;
    const int e = idx >> 3;       // edge id
    const int t = idx & 7;        // time step
    if (e >= N_EDGES) return;

    const int s = src[e];
    const int d = tgt[e];
    const float4* __restrict__ xj = (const float4*)(hu + (size_t)s * (T_DIM * C_DIM) + t * C_DIM);
    const float4* __restrict__ xi = (const float4*)(hu + (size_t)d * (T_DIM * C_DIM) + t * C_DIM);

    float4 j[8];
    float dot = 0.0f;
#pragma unroll
    for (int k = 0; k < 8; ++k) {
        j[k] = xj[k];
        const float4 i4 = xi[k];
        dot += j[k].x * i4.x + j[k].y * i4.y + j[k].z * i4.z + j[k].w * i4.w;
    }
    const float gate = 1.0f / (1.0f + __expf(-dot));

    unsigned* __restrict__ out = agg + (size_t)d * (T_DIM * C_DIM) + t * C_DIM;
#pragma unroll
    for (int k = 0; k < 8; ++k) {
        atomicMax(out + 4 * k + 0, enc_f32(j[k].x * gate));
        atomicMax(out + 4 * k + 1, enc_f32(j[k].y * gate));
        atomicMax(out + 4 * k + 2, enc_f32(j[k].z * gate));
        atomicMax(out + 4 * k + 3, enc_f32(j[k].w * gate));
    }
}

// out = leaky_relu(h + (finite ? decode(agg) : 0), 0.01)
__global__ void gat_combine(const float* __restrict__ h,
                            const unsigned* __restrict__ agg,
                            float* __restrict__ out, int n) {
    int i = blockIdx.x * blockDim.x + threadIdx.x;
    if (i >= n) return;
    const unsigned u = agg[i];
    const unsigned b = (u & 0x80000000u) ? (u & 0x7FFFFFFFu) : ~u;
    float a = __uint_as_float(b);
    if ((b & 0x7F800000u) == 0x7F800000u) a = 0.0f;   // NaN/Inf (incl. empty) -> 0
    const float v = h[i] + a;
    out[i] = (v >= 0.0f) ? v : 0.01f * v;
}

// ---------------------------------------------------------------------------
extern "C" void kernel_launch(void* const* d_in, const int* in_sizes, int n_in,
                              void* d_out, int out_size, void* d_ws, size_t ws_size,
                              hipStream_t stream) {
    const float* X   = (const float*)d_in[0];
    const int*   ei  = (const int*)d_in[1];
    /* d_in[2] = edge_attr, d_in[4] = We1, d_in[7] = We2 : dead code in reference */
    const float* Wn1 = (const float*)d_in[3];
    const float* u1  = (const float*)d_in[5];
    const float* Wn2 = (const float*)d_in[6];
    const float* u2  = (const float*)d_in[8];
    float* out = (float*)d_out;

    float*    h   = (float*)d_ws;           // [M_ROWS, 32]
    float*    hu  = h + NTC;                // [M_ROWS, 32]
    unsigned* agg = (unsigned*)(hu + NTC);  // [M_ROWS, 32] encoded max

    const int* src = ei;
    const int* tgt = ei + N_EDGES;

    const dim3 gemm_grid((M_ROWS / 16 + 3) / 4);
    const dim3 gemm_blk(128);
    const int  n_elem   = NTC;
    const dim3 elem_grid((n_elem + 255) / 256);
    const dim3 edge_grid((N_EDGES * T_DIM + 255) / 256);
    const dim3 blk256(256);

    // ----- layer 1 -----
    gat_gemm32<<<gemm_grid, gemm_blk, 0, stream>>>(X, Wn1, h, M_ROWS);
    gat_gemm32<<<gemm_grid, gemm_blk, 0, stream>>>(h, u1, hu, M_ROWS);
    gat_init_agg<<<elem_grid, blk256, 0, stream>>>(agg, n_elem);
    gat_edge<<<edge_grid, blk256, 0, stream>>>(hu, src, tgt, agg);
    gat_combine<<<elem_grid, blk256, 0, stream>>>(h, agg, out, n_elem);  // c1 -> d_out

    // ----- layer 2 (reads c1 from d_out, final result overwrites d_out) -----
    gat_gemm32<<<gemm_grid, gemm_blk, 0, stream>>>(out, Wn2, h, M_ROWS);
    gat_gemm32<<<gemm_grid, gemm_blk, 0, stream>>>(h, u2, hu, M_ROWS);
    gat_init_agg<<<elem_grid, blk256, 0, stream>>>(agg, n_elem);
    gat_edge<<<edge_grid, blk256, 0, stream>>>(hu, src, tgt, agg);
    gat_combine<<<elem_grid, blk256, 0, stream>>>(h, agg, out, n_elem);
}